// Semantic_81097572483746
// MI455X (gfx1250) — compile-verified
//
#include <hip/hip_runtime.h>

// ---------------------------------------------------------------------------
// MI455X (gfx1250, wave32) implementation.
// Memory-bound problem (~640MB traffic -> ~27us floor @ 23.3TB/s), ~100 GFLOP
// of GEMM work done with V_WMMA_F32_16X16X4_F32 (full f32 precision).
// Global->LDS staging uses CDNA5 async-to-LDS ops (ASYNCcnt tracked).
// ---------------------------------------------------------------------------

typedef __attribute__((ext_vector_type(2))) float v2f;
typedef __attribute__((ext_vector_type(4))) float v4f;
typedef __attribute__((ext_vector_type(8))) float v8f;

#define FD   128     // feature dim
#define NN   16      // nodes per (b,s)
#define EE   256     // edges per (b,s)
#define PROJ 300     // projection dim
#define PTILES 19    // ceil(300/16)
#define WROW 132     // LDS stride for 128-col tiles (bank-conflict-free: 132%64=4)
#define TROW 18      // LDS stride for 16-col tiles  (even -> 8B aligned, 18r%64 distinct)

// LDS layout (float offsets)
#define OFF_E    0
#define OFF_N    (OFF_E   + EE*WROW)      // edges 256x132
#define OFF_W1   (OFF_N   + NN*WROW)      // nodes 16x132
#define OFF_W2   (OFF_W1  + 16*WROW)
#define OFF_W3   (OFF_W2  + 16*WROW)
#define OFF_F1T  (OFF_W3  + 16*WROW)      // current f1 p-tile 16x18
#define OFF_F3T  (OFF_F1T + 16*TROW)      // current f3 p-tile 16x18
#define OFF_HH   (OFF_F3T + 16*TROW)      // per-wave h transpose buf 8 x 16x18
#define OFF_MSQH (OFF_HH  + 8*16*TROW)    // 256
#define OFF_MSQ1 (OFF_MSQH + EE)          // 16
#define OFF_REDF (OFF_MSQ1 + NN)          // 8
#define OFF_REDU (OFF_REDF + 8)           // 8
#define LDS_FLOATS (OFF_REDU + 8)         // = 45408 floats = 181632 bytes

static __device__ __forceinline__ v8f wmma4(v2f a, v2f b, v8f c) {
    // D = A(16x4 f32) x B(4x16 f32) + C(16x16 f32)
    return __builtin_amdgcn_wmma_f32_16x16x4_f32(false, a, false, b, (short)0, c,
                                                 false, false);
}

// CDNA5 async copy: each lane copies 16 bytes from the global address in gaddr to the LDS offset in lds_off.
// Tracked by ASYNCcnt (not LOADcnt); completion via s_wait_asynccnt.
static __device__ __forceinline__ void async_g2l_b128(unsigned lds_off,
                                                      unsigned long long gaddr) {
    asm volatile("global_load_async_to_lds_b128 %0, %1, off"
                 :: "v"(lds_off), "v"(gaddr) : "memory");
}
static __device__ __forceinline__ void async_wait0() {
    asm volatile("s_wait_asynccnt 0x0" ::: "memory");
}

__global__ void __launch_bounds__(256)
graph_l2_kernel(const float* __restrict__ nodes, const float* __restrict__ edges,
                const int* __restrict__ adj,     const int* __restrict__ tlen,
                const float* __restrict__ W1, const float* __restrict__ b1,
                const float* __restrict__ W2, const float* __restrict__ b2,
                const float* __restrict__ W3, const float* __restrict__ b3,
                float* __restrict__ ws)
{
    extern __shared__ float sm[];
    const int q   = blockIdx.x;      // 0..4095, q = b*16 + s
    const int bb  = q >> 4;
    const int ssq = q & 15;
    if (ssq >= tlen[bb]) return;     // masked-out sequence step: no contribution

    const int tid  = threadIdx.x;
    const int lane = tid & 31;
    const int wv   = tid >> 5;       // 8 waves
    const int hl   = lane >> 4;      // half-wave select
    const int l16  = lane & 15;

    float* sE   = sm + OFF_E;
    float* sN   = sm + OFF_N;
    float* sW1  = sm + OFF_W1;
    float* sW2  = sm + OFF_W2;
    float* sW3  = sm + OFF_W3;
    float* sF1T = sm + OFF_F1T;
    float* sF3T = sm + OFF_F3T;
    float* sHB  = sm + OFF_HH + wv * 16 * TROW;   // per-wave transpose buffer
    float* sMsqh = sm + OFF_MSQH;
    float* sMsq1 = sm + OFF_MSQ1;
    float* sRedF = sm + OFF_REDF;
    unsigned* sRedU = (unsigned*)(sm + OFF_REDU);

    // LDS byte offsets (generic LDS address low 32 bits == wave LDS offset)
    const unsigned sE_b  = (unsigned)(uintptr_t)sE;
    const unsigned sN_b  = (unsigned)(uintptr_t)sN;
    const unsigned sW1_b = (unsigned)(uintptr_t)sW1;
    const unsigned sW2_b = (unsigned)(uintptr_t)sW2;
    const unsigned sW3_b = (unsigned)(uintptr_t)sW3;

    // ---- stage edges (256x128) + nodes (16x128) via async-to-LDS b128 ----
    const unsigned long long egb =
        (unsigned long long)(uintptr_t)(edges + (size_t)q * EE * FD);
    #pragma unroll 8
    for (int j = 0; j < 32; ++j) {              // 8192 float4 / 256 threads
        int idx = tid + 256 * j;
        int e = idx >> 5, f4 = idx & 31;
        async_g2l_b128(sE_b + (unsigned)(e * WROW + f4 * 4) * 4u,
                       egb + (unsigned long long)idx * 16u);
    }
    const unsigned long long ngb =
        (unsigned long long)(uintptr_t)(nodes + (size_t)q * NN * FD);
    #pragma unroll
    for (int j = 0; j < 2; ++j) {
        int idx = tid + 256 * j;
        int n = idx >> 5, f4 = idx & 31;
        async_g2l_b128(sN_b + (unsigned)(n * WROW + f4 * 4) * 4u,
                       ngb + (unsigned long long)idx * 16u);
    }
    async_wait0();

    const unsigned long long w1b = (unsigned long long)(uintptr_t)W1;
    const unsigned long long w2b = (unsigned long long)(uintptr_t)W2;
    const unsigned long long w3b = (unsigned long long)(uintptr_t)W3;

    v8f   Dacc[2]  = {};             // D = f1 . h^T accumulators (2 e-tiles)
    float msqh[2][8] = {};           // per-lane sum of h^2 over this lane's cols
    float msq1a[8]   = {};           // wave0 only

    for (int pt = 0; pt < PTILES; ++pt) {
        const int p0 = pt * 16;
        __syncthreads();
        // ---- stage 16-row slices of W1/W2/W3 ----
        if (p0 + 16 <= PROJ) {       // full tile: async-to-LDS (block-uniform)
            #pragma unroll
            for (int j = 0; j < 2; ++j) {
                int idx = tid + 256 * j;             // 0..511 float4
                int pp = idx >> 5, f4 = idx & 31;
                unsigned lofs = (unsigned)((pp * WROW + f4 * 4) * 4);
                unsigned long long gofs =
                    (unsigned long long)((p0 + pp) * FD + f4 * 4) * 4u;
                async_g2l_b128(sW1_b + lofs, w1b + gofs);
                async_g2l_b128(sW2_b + lofs, w2b + gofs);
                async_g2l_b128(sW3_b + lofs, w3b + gofs);
            }
            async_wait0();
        } else {                     // last tile: zero-pad rows >= 300
            #pragma unroll
            for (int j = 0; j < 2; ++j) {
                int idx = tid + 256 * j;
                int pp = idx >> 5, f4 = idx & 31;
                int prow = p0 + pp;
                v4f z = {0.f, 0.f, 0.f, 0.f};
                v4f w1v = z, w2v = z, w3v = z;
                if (prow < PROJ) {
                    w1v = *(const v4f*)(W1 + prow * FD + f4 * 4);
                    w2v = *(const v4f*)(W2 + prow * FD + f4 * 4);
                    w3v = *(const v4f*)(W3 + prow * FD + f4 * 4);
                }
                *(v4f*)(sW1 + pp * WROW + f4 * 4) = w1v;
                *(v4f*)(sW2 + pp * WROW + f4 * 4) = w2v;
                *(v4f*)(sW3 + pp * WROW + f4 * 4) = w3v;
            }
        }
        __syncthreads();

        const int pcol = p0 + l16;               // this lane's P column

        // ---- wave0: f1 tile, wave1: f3 tile (16x16 each, K=128) ----
        if (wv < 2) {
            const float* sWx = wv ? sW3 : sW1;
            const float* bx  = wv ? b3  : b1;
            v8f acc = {};
            #pragma unroll
            for (int kk = 0; kk < 32; ++kk) {
                int k0 = kk * 4 + 2 * hl;
                v2f a = *(const v2f*)(sN  + l16 * WROW + k0);   // A: nodes rows
                v2f b = *(const v2f*)(sWx + l16 * WROW + k0);   // B: W^T cols
                acc = wmma4(a, b, acc);
            }
            float bv = (pcol < PROJ) ? bx[pcol] : 0.f;
            float* dst = wv ? sF3T : sF1T;
            #pragma unroll
            for (int r = 0; r < 8; ++r) {
                float v = acc[r] + bv;
                dst[(r + 8 * hl) * TROW + l16] = v;             // C/D layout store
                if (wv == 0) msq1a[r] += v * v;
            }
        }

        // ---- all waves: f2 tiles for e-tiles {wv, wv+8} (shared B frags) ----
        v8f f2f[2] = {};
        {
            const float* eA0 = sE + (wv * 16 + l16) * WROW;
            const float* eA1 = sE + ((wv + 8) * 16 + l16) * WROW;
            #pragma unroll
            for (int kk = 0; kk < 32; ++kk) {
                int k0 = kk * 4 + 2 * hl;
                v2f b  = *(const v2f*)(sW2 + l16 * WROW + k0);
                v2f a0 = *(const v2f*)(eA0 + k0);
                v2f a1 = *(const v2f*)(eA1 + k0);
                f2f[0] = wmma4(a0, b, f2f[0]);
                f2f[1] = wmma4(a1, b, f2f[1]);
            }
            float bv = (pcol < PROJ) ? b2[pcol] : 0.f;
            #pragma unroll
            for (int i = 0; i < 2; ++i)
                #pragma unroll
                for (int r = 0; r < 8; ++r) f2f[i][r] += bv;
        }
        __syncthreads();   // f1/f3 tiles visible to all waves

        // ---- hoisted fragments shared by both e-tiles ----
        v8f f3frag;
        #pragma unroll
        for (int r = 0; r < 8; ++r)
            f3frag[r] = sF3T[(r + 8 * hl) * TROW + l16];
        v2f aD[4];
        #pragma unroll
        for (int kk = 0; kk < 4; ++kk)
            aD[kk] = *(const v2f*)(sF1T + l16 * TROW + kk * 4 + 2 * hl);

        // ---- h = f2 - f3 (e%16 == local row!), per-lane msqh, D-GEMM ----
        #pragma unroll
        for (int i = 0; i < 2; ++i) {
            #pragma unroll
            for (int r = 0; r < 8; ++r) {
                float hv = f2f[i][r] - f3frag[r];
                msqh[i][r] += hv * hv;                       // defer reduction
                sHB[(r + 8 * hl) * TROW + l16] = hv;         // stage transpose
            }
            #pragma unroll
            for (int kk = 0; kk < 4; ++kk) {                 // K=16 this p-tile
                v2f b = *(const v2f*)(sHB + l16 * TROW + kk * 4 + 2 * hl);
                Dacc[i] = wmma4(aD[kk], b, Dacc[i]);
            }
        }
    }

    // ---- one-time cross-lane reductions (16-lane halves) ----
    #pragma unroll
    for (int i = 0; i < 2; ++i) {
        int et = wv + 8 * i;
        #pragma unroll
        for (int r = 0; r < 8; ++r) {
            float sq = msqh[i][r];
            sq += __shfl_xor(sq, 1);
            sq += __shfl_xor(sq, 2);
            sq += __shfl_xor(sq, 4);
            sq += __shfl_xor(sq, 8);
            if (l16 == 0) sMsqh[et * 16 + r + 8 * hl] = sq;
        }
    }
    if (wv == 0) {
        #pragma unroll
        for (int r = 0; r < 8; ++r) {
            float sq = msq1a[r];
            sq += __shfl_xor(sq, 1);
            sq += __shfl_xor(sq, 2);
            sq += __shfl_xor(sq, 4);
            sq += __shfl_xor(sq, 8);
            if (l16 == 0) sMsq1[r + 8 * hl] = sq;
        }
    }
    __syncthreads();

    // ---- l = (msq1[n] + msqh[e] + 2*D[n,e]) / 300, mask, accumulate ----
    const int* adjq = adj + (size_t)q * NN * EE;
    float tsum = 0.f; unsigned tcnt = 0;
    #pragma unroll
    for (int i = 0; i < 2; ++i) {
        int e = (wv + 8 * i) * 16 + l16;
        float mh = sMsqh[e];
        #pragma unroll
        for (int r = 0; r < 8; ++r) {
            int n = r + 8 * hl;
            float lv = (sMsq1[n] + mh + 2.f * Dacc[i][r]) * (1.0f / 300.0f);
            if (adjq[n * EE + e] == 1) {
                tsum += lv;
                if (lv > 0.f) tcnt++;
            }
        }
    }
    tsum += __shfl_xor(tsum, 1);  tcnt += (unsigned)__shfl_xor((int)tcnt, 1);
    tsum += __shfl_xor(tsum, 2);  tcnt += (unsigned)__shfl_xor((int)tcnt, 2);
    tsum += __shfl_xor(tsum, 4);  tcnt += (unsigned)__shfl_xor((int)tcnt, 4);
    tsum += __shfl_xor(tsum, 8);  tcnt += (unsigned)__shfl_xor((int)tcnt, 8);
    tsum += __shfl_xor(tsum, 16); tcnt += (unsigned)__shfl_xor((int)tcnt, 16);
    if (lane == 0) { sRedF[wv] = tsum; sRedU[wv] = tcnt; }
    __syncthreads();
    if (tid == 0) {
        float t = 0.f; unsigned c = 0;
        #pragma unroll
        for (int w = 0; w < 8; ++w) { t += sRedF[w]; c += sRedU[w]; }
        atomicAdd(ws, t);
        atomicAdd((unsigned*)ws + 1, c);
    }
}

__global__ void init_ws_kernel(float* ws) {
    ws[0] = 0.f;
    ((unsigned*)ws)[1] = 0u;
}

__global__ void finish_kernel(const float* ws, float* out) {
    unsigned c = ((const unsigned*)ws)[1];
    out[0] = ws[0] / (float)c;
}

extern "C" void kernel_launch(void* const* d_in, const int* in_sizes, int n_in,
                              void* d_out, int out_size, void* d_ws, size_t ws_size,
                              hipStream_t stream) {
    (void)in_sizes; (void)n_in; (void)out_size; (void)ws_size;
    const float* nodes = (const float*)d_in[0];
    const float* edges = (const float*)d_in[1];
    const int*   adjm  = (const int*)d_in[2];
    const int*   tlen  = (const int*)d_in[3];
    const float* W1 = (const float*)d_in[4];
    const float* b1 = (const float*)d_in[5];
    const float* W2 = (const float*)d_in[6];
    const float* b2 = (const float*)d_in[7];
    const float* W3 = (const float*)d_in[8];
    const float* b3 = (const float*)d_in[9];
    float* ws = (float*)d_ws;

    const int smem_bytes = LDS_FLOATS * 4;   // ~178 KB of the WGP's 320 KB LDS
    hipFuncSetAttribute(reinterpret_cast<const void*>(graph_l2_kernel),
                        hipFuncAttributeMaxDynamicSharedMemorySize, smem_bytes);

    init_ws_kernel<<<1, 1, 0, stream>>>(ws);
    graph_l2_kernel<<<4096, 256, smem_bytes, stream>>>(
        nodes, edges, adjm, tlen, W1, b1, W2, b2, W3, b3, ws);
    finish_kernel<<<1, 1, 0, stream>>>(ws, (float*)d_out);
}